// DeepKalmanFilter_2774548873355
// MI455X (gfx1250) — compile-verified
//
#include <hip/hip_runtime.h>

// DeepKalmanFilter fused kernel for gfx1250 (MI455X).
// One wave (32 lanes) owns a 16-row batch tile for the entire T=512 double scan.
// All GEMMs are V_WMMA_F32_16X16X4_F32 with weights pre-staged in A-layout regs.
// Per-step global loads are software-pipelined one iteration ahead; pointwise
// nonlinearities use the CDNA5 hardware v_tanh_f32; streaming traffic is
// non-temporal to keep L2 for the h_seq spill.

typedef float v2f __attribute__((ext_vector_type(2)));
typedef float v4f __attribute__((ext_vector_type(4)));
typedef float v8f __attribute__((ext_vector_type(8)));

#define T_DIM 512
#define B_DIM 8192
#define NTILE (B_DIM / 16)   // 512 batch tiles
#define WPB   8              // waves per block

__device__ __forceinline__ float hswap(float v) {
  // swap halves of the wave32 (lane L <-> lane L^16)
  return __shfl_xor(v, 16, 32);
}
__device__ __forceinline__ float tanh_hw(float x) {
  float r;
  asm("v_tanh_f32 %0, %1" : "=v"(r) : "v"(x));
  return r;
}
__device__ __forceinline__ float sigm(float x) {
  // sigmoid(x) = 0.5*tanh(x/2) + 0.5
  return __builtin_fmaf(0.5f, tanh_hw(0.5f * x), 0.5f);
}

__device__ __forceinline__ v8f wmma4(v2f a, v2f b, v8f c) {
  // D(16x16,f32) = A(16x4,f32) * B(4x16,f32) + C
  return __builtin_amdgcn_wmma_f32_16x16x4_f32(false, a, false, b, (short)0, c, false, false);
}

// Build the 4 B-operand K-chunks (K=16 -> 4x K=4) of a 16x16 matrix currently
// held in C/D layout registers S[0..7] (lane L, reg r <-> M = r+8*(L>=16), N = L%16).
// B-layout chunk kk, vgpr j, lane L <-> K = 4*kk + 2*(L>=16) + j, N = L%16.
__device__ __forceinline__ void make_b(const float* S, bool lo, v2f* b) {
  float w0 = hswap(S[0]), w1 = hswap(S[1]), w2 = hswap(S[2]), w3 = hswap(S[3]);
  float w4 = hswap(S[4]), w5 = hswap(S[5]), w6 = hswap(S[6]), w7 = hswap(S[7]);
  b[0][0] = lo ? S[0] : w2;  b[0][1] = lo ? S[1] : w3;
  b[1][0] = lo ? S[4] : w6;  b[1][1] = lo ? S[5] : w7;
  b[2][0] = lo ? w0 : S[2];  b[2][1] = lo ? w1 : S[3];
  b[3][0] = lo ? w4 : S[6];  b[3][1] = lo ? w5 : S[7];
}

__global__ __launch_bounds__(WPB * 32)
void dkf_kernel(const float* __restrict__ x,    const float* __restrict__ eps,
                const float* __restrict__ W_ih, const float* __restrict__ W_hh,
                const float* __restrict__ b_ih, const float* __restrict__ b_hh,
                const float* __restrict__ Wc1,  const float* __restrict__ bc1,
                const float* __restrict__ Wc2,  const float* __restrict__ bc2,
                const float* __restrict__ We1,  const float* __restrict__ be1,
                const float* __restrict__ We2,  const float* __restrict__ be2,
                const float* __restrict__ Wd1,  const float* __restrict__ bd1,
                const float* __restrict__ Wd2,  const float* __restrict__ bd2,
                float* __restrict__ out) {
  const int lane = threadIdx.x & 31;
  const int wid  = threadIdx.x >> 5;
  const int tile = blockIdx.x * WPB + wid;       // 0..511
  const int row  = lane & 15;                    // batch row within tile
  const int hi   = lane >> 4;                    // wave half
  const bool lo  = (hi == 0);
  const int r0   = tile * 16;

  const size_t TB = (size_t)T_DIM * B_DIM;
  float* out_mux = out;                // (T,B)
  float* out_lvx = out + TB;           // (T,B)
  float* out_muz = out + 2 * TB;       // (T,B,16)  -- also h_seq scratch
  float* out_lvz = out + 18 * TB;      // (T,B,16)
  float* out_z   = out + 34 * TB;      // (T,B,16)

  const v8f z8 = {0.f, 0.f, 0.f, 0.f, 0.f, 0.f, 0.f, 0.f};

  // ---------------- Phase 1: reverse LSTM ----------------
  // A' = [W_hh | W_ih | (b_ih+b_hh) | 0 | 0]  (64 x 20), A-layout, constant.
  v2f a1[4][5];
#pragma unroll
  for (int tg = 0; tg < 4; ++tg) {
    const int g = tg * 16 + row;
#pragma unroll
    for (int kk = 0; kk < 5; ++kk) {
#pragma unroll
      for (int j = 0; j < 2; ++j) {
        const int k = kk * 4 + hi * 2 + j;
        float v;
        if (k < 16)       v = W_hh[g * 16 + k];
        else if (k == 16) v = W_ih[g];
        else if (k == 17) v = b_ih[g] + b_hh[g];
        else              v = 0.f;
        a1[tg][kk][j] = v;
      }
    }
  }

  float hT[8], cT[8];
#pragma unroll
  for (int r = 0; r < 8; ++r) { hT[r] = 0.f; cT[r] = 0.f; }

  // x is streamed once: non-temporal, prefetched one step ahead.
  float xv = __builtin_nontemporal_load(x + (size_t)(T_DIM - 1) * B_DIM + r0 + row);
  for (int t = T_DIM - 1; t >= 0; --t) {
    const int tp = (t > 0) ? (t - 1) : 0;
    const float xv_n = __builtin_nontemporal_load(x + (size_t)tp * B_DIM + r0 + row);

    v2f b[5];
    make_b(hT, lo, b);                                   // hT chunks
    b[4][0] = lo ? xv : 0.f;                             // k=16 -> x, k=18 -> 0
    b[4][1] = lo ? 1.f : 0.f;                            // k=17 -> bias row

    v8f acc[4];
#pragma unroll
    for (int tg = 0; tg < 4; ++tg) {
      acc[tg] = z8;
#pragma unroll
      for (int kk = 0; kk < 5; ++kk) acc[tg] = wmma4(a1[tg][kk], b[kk], acc[tg]);
    }
    // gates^T tiles: 0=i, 1=f, 2=g, 3=o (PyTorch order)
#pragma unroll
    for (int r = 0; r < 8; ++r) {
      const float ig = sigm(acc[0][r]);
      const float fg = sigm(acc[1][r]);
      const float gg = tanh_hw(acc[2][r]);
      const float og = sigm(acc[3][r]);
      cT[r] = fg * cT[r] + ig * gg;
      hT[r] = og * tanh_hw(cT[r]);
    }
    // Spill h_t into the mu_z output block for (t, tile) (overwritten in phase 2).
    // Regular (RT) stores: phase 2 wants these to hit L2.
    float* sp = out_muz + (((size_t)t * NTILE + tile) * 32 + lane) * 8;
    v4f s0 = {hT[0], hT[1], hT[2], hT[3]};
    v4f s1 = {hT[4], hT[5], hT[6], hT[7]};
    *(v4f*)(sp)     = s0;
    *(v4f*)(sp + 4) = s1;

    xv = xv_n;
  }

  // ---------------- Phase 2 constants ----------------
  // A_c1 = [Wc1 | bc1 | 0]  (16 x 36): cols 0..15 hit h, 16..31 hit z.
  v2f ac1[9];
#pragma unroll
  for (int kk = 0; kk < 9; ++kk)
#pragma unroll
    for (int j = 0; j < 2; ++j) {
      const int k = kk * 4 + hi * 2 + j;
      ac1[kk][j] = (k < 32) ? Wc1[row * 32 + k] : ((k == 32) ? bc1[row] : 0.f);
    }
  // Compose the two bias-linear stages (no nonlinearity between them):
  // e1pre = We1 @ (Wc2 @ c1 + bc2) + be1 = (We1@Wc2) @ c1 + (We1@bc2 + be1)
  v2f ace[5];
#pragma unroll
  for (int kk = 0; kk < 5; ++kk)
#pragma unroll
    for (int j = 0; j < 2; ++j) {
      const int k = kk * 4 + hi * 2 + j;
      float v = 0.f;
      if (k < 16) {
        for (int q = 0; q < 8; ++q) v += We1[row * 8 + q] * Wc2[q * 16 + k];
      } else if (k == 16) {
        v = be1[row];
        for (int q = 0; q < 8; ++q) v += We1[row * 8 + q] * bc2[q];
      }
      ace[kk][j] = v;
    }
  // A_e2 = [We2 | be2 | 0]  (32 x 20): tile 0 -> mu_z rows, tile 1 -> logvar rows
  v2f ae2[2][5];
#pragma unroll
  for (int tg = 0; tg < 2; ++tg) {
    const int m = tg * 16 + row;
#pragma unroll
    for (int kk = 0; kk < 5; ++kk)
#pragma unroll
      for (int j = 0; j < 2; ++j) {
        const int k = kk * 4 + hi * 2 + j;
        ae2[tg][kk][j] = (k < 16) ? We2[m * 16 + k] : ((k == 16) ? be2[m] : 0.f);
      }
  }
  // A_d1 = [Wd1 | bd1 | 0]  (16 x 20)
  v2f ad1[5];
#pragma unroll
  for (int kk = 0; kk < 5; ++kk)
#pragma unroll
    for (int j = 0; j < 2; ++j) {
      const int k = kk * 4 + hi * 2 + j;
      ad1[kk][j] = (k < 16) ? Wd1[row * 16 + k] : ((k == 16) ? bd1[row] : 0.f);
    }
  float wd2a[8], wd2b[8];
#pragma unroll
  for (int r = 0; r < 8; ++r) {
    wd2a[r] = Wd2[r + 8 * hi];
    wd2b[r] = Wd2[16 + r + 8 * hi];
  }
  const float bd20 = bd2[0], bd21 = bd2[1];

  // ---------------- Phase 2: z-scan + decoder (software-pipelined) ----------
  float zT[8];
#pragma unroll
  for (int r = 0; r < 8; ++r) zT[r] = 0.f;

  v2f bb; bb[0] = lo ? 1.f : 0.f; bb[1] = 0.f;   // bias row chunk

  // z_prev(t=0) == 0 -> its B chunks are zero.
  v2f zb[4];
#pragma unroll
  for (int kk = 0; kk < 4; ++kk) { zb[kk][0] = 0.f; zb[kk][1] = 0.f; }

  // Prologue: load h(0), eps(0) and pre-swizzle h.
  v2f bh[4];
  float er[8];
  {
    const float* sp0 = out_muz + (((size_t)0 * NTILE + tile) * 32 + lane) * 8;
    const v4f h0 = *(const v4f*)(sp0);
    const v4f h1 = *(const v4f*)(sp0 + 4);
    const float hR[8] = {h0[0], h0[1], h0[2], h0[3], h1[0], h1[1], h1[2], h1[3]};
    make_b(hR, lo, bh);
    const float* ep0 = eps + ((size_t)0 * B_DIM + r0 + row) * 16 + 8 * hi;
    const v4f e0 = __builtin_nontemporal_load((const v4f*)ep0);
    const v4f e1 = __builtin_nontemporal_load((const v4f*)(ep0 + 4));
#pragma unroll
    for (int r = 0; r < 4; ++r) { er[r] = e0[r]; er[4 + r] = e1[r]; }
  }

  for (int t = 0; t < T_DIM; ++t) {
    // Prefetch next step's h and eps right away (hidden under this step).
    const int tn = (t + 1 < T_DIM) ? (t + 1) : t;
    const float* spn = out_muz + (((size_t)tn * NTILE + tile) * 32 + lane) * 8;
    const v4f hna = *(const v4f*)(spn);
    const v4f hnb = *(const v4f*)(spn + 4);
    const float* epn = eps + ((size_t)tn * B_DIM + r0 + row) * 16 + 8 * hi;
    const v4f ena = __builtin_nontemporal_load((const v4f*)epn);
    const v4f enb = __builtin_nontemporal_load((const v4f*)(epn + 4));

    // GEMM1: c1 = Wc1 @ [h; z] + bc1  (h chunks first: independent of z(t-1))
    v8f c1 = z8;
#pragma unroll
    for (int kk = 0; kk < 4; ++kk) c1 = wmma4(ac1[kk],     bh[kk], c1);
#pragma unroll
    for (int kk = 0; kk < 4; ++kk) c1 = wmma4(ac1[4 + kk], zb[kk], c1);
    c1 = wmma4(ac1[8], bb, c1);

    float s1[8];
#pragma unroll
    for (int r = 0; r < 8; ++r) s1[r] = tanh_hw(c1[r]);

    v2f b1[4]; make_b(s1, lo, b1);
    v8f e1 = z8;
#pragma unroll
    for (int kk = 0; kk < 4; ++kk) e1 = wmma4(ace[kk], b1[kk], e1);
    e1 = wmma4(ace[4], bb, e1);

    float s2[8];
#pragma unroll
    for (int r = 0; r < 8; ++r) s2[r] = tanh_hw(e1[r]);

    v2f b2[4]; make_b(s2, lo, b2);
    v8f mu = z8, lv = z8;
#pragma unroll
    for (int kk = 0; kk < 4; ++kk) mu = wmma4(ae2[0][kk], b2[kk], mu);
    mu = wmma4(ae2[0][4], bb, mu);
#pragma unroll
    for (int kk = 0; kk < 4; ++kk) lv = wmma4(ae2[1][kk], b2[kk], lv);
    lv = wmma4(ae2[1][4], bb, lv);

    // z = mu + eps * exp(0.5*logvar)
#pragma unroll
    for (int r = 0; r < 8; ++r) zT[r] = mu[r] + er[r] * __expf(0.5f * lv[r]);

    // Swizzle z once: feeds both the decoder and next step's GEMM1.
    make_b(zT, lo, zb);

    // Stores: none of these are re-read -> non-temporal (keep L2 for h_seq).
    const size_t ob = ((size_t)t * B_DIM + r0 + row) * 16 + 8 * hi;
    {
      v4f m0 = {mu[0], mu[1], mu[2], mu[3]}, m1 = {mu[4], mu[5], mu[6], mu[7]};
      v4f l0 = {lv[0], lv[1], lv[2], lv[3]}, l1 = {lv[4], lv[5], lv[6], lv[7]};
      v4f q0 = {zT[0], zT[1], zT[2], zT[3]}, q1 = {zT[4], zT[5], zT[6], zT[7]};
      __builtin_nontemporal_store(m0, (v4f*)(out_muz + ob));
      __builtin_nontemporal_store(m1, (v4f*)(out_muz + ob + 4));
      __builtin_nontemporal_store(l0, (v4f*)(out_lvz + ob));
      __builtin_nontemporal_store(l1, (v4f*)(out_lvz + ob + 4));
      __builtin_nontemporal_store(q0, (v4f*)(out_z + ob));
      __builtin_nontemporal_store(q1, (v4f*)(out_z + ob + 4));
    }

    // Fused decoder: d1 = tanh(Wd1 @ z + bd1); (mu_x, logvar_x) = Wd2 @ d1 + bd2
    v8f d1 = z8;
#pragma unroll
    for (int kk = 0; kk < 4; ++kk) d1 = wmma4(ad1[kk], zb[kk], d1);
    d1 = wmma4(ad1[4], bb, d1);

    float p0 = 0.f, p1 = 0.f;
#pragma unroll
    for (int r = 0; r < 8; ++r) {
      const float tt = tanh_hw(d1[r]);
      p0 += wd2a[r] * tt;
      p1 += wd2b[r] * tt;
    }
    const float m0 = p0 + hswap(p0) + bd20;
    const float l0 = p1 + hswap(p1) + bd21;
    if (lo) {
      __builtin_nontemporal_store(m0, out_mux + (size_t)t * B_DIM + r0 + row);
      __builtin_nontemporal_store(l0, out_lvx + (size_t)t * B_DIM + r0 + row);
    }

    // Rotate prefetched h/eps into place for the next iteration.
    const float hRn[8] = {hna[0], hna[1], hna[2], hna[3], hnb[0], hnb[1], hnb[2], hnb[3]};
    make_b(hRn, lo, bh);
#pragma unroll
    for (int r = 0; r < 4; ++r) { er[r] = ena[r]; er[4 + r] = enb[r]; }
  }
}

extern "C" void kernel_launch(void* const* d_in, const int* in_sizes, int n_in,
                              void* d_out, int out_size, void* d_ws, size_t ws_size,
                              hipStream_t stream) {
  (void)in_sizes; (void)n_in; (void)out_size; (void)d_ws; (void)ws_size;
  const float* X    = (const float*)d_in[0];
  const float* EPS  = (const float*)d_in[1];
  const float* W_ih = (const float*)d_in[2];
  const float* W_hh = (const float*)d_in[3];
  const float* b_ih = (const float*)d_in[4];
  const float* b_hh = (const float*)d_in[5];
  const float* Wc1  = (const float*)d_in[6];
  const float* bc1  = (const float*)d_in[7];
  const float* Wc2  = (const float*)d_in[8];
  const float* bc2  = (const float*)d_in[9];
  const float* We1  = (const float*)d_in[10];
  const float* be1  = (const float*)d_in[11];
  const float* We2  = (const float*)d_in[12];
  const float* be2  = (const float*)d_in[13];
  const float* Wd1  = (const float*)d_in[14];
  const float* bd1  = (const float*)d_in[15];
  const float* Wd2  = (const float*)d_in[16];
  const float* bd2  = (const float*)d_in[17];

  dim3 grid(NTILE / WPB);   // 64 blocks
  dim3 block(WPB * 32);     // 8 waves of 32
  dkf_kernel<<<grid, block, 0, stream>>>(X, EPS, W_ih, W_hh, b_ih, b_hh,
                                         Wc1, bc1, Wc2, bc2, We1, be1,
                                         We2, be2, Wd1, bd1, Wd2, bd2,
                                         (float*)d_out);
}